// AdderLayer_16209206575289
// MI455X (gfx1250) — compile-verified
//
#include <hip/hip_runtime.h>
#include <cmath>

// Problem constants (match reference)
#define NN   8
#define CIN  64
#define COUT 64
#define HH   128
#define WW   128
#define TILE 16
#define HALO 18                      // TILE + 2 (3x3, pad 1)
#define NPIX (NN*COUT*HH*WW)         // 8,388,608 elements
#define NBLK1 (NN*(HH/TILE)*(WW/TILE))  // 512 blocks in stage 1
#define CNTF ((float)(NN*HH*WW))     // 131072 elements per channel for BN stats
#define WCO  (CIN*9)                 // weight stride per output channel = 576

typedef float v2f __attribute__((ext_vector_type(2)));
typedef float v8f __attribute__((ext_vector_type(8)));
typedef __attribute__((address_space(3))) float lds_float;

// ---------------------------------------------------------------------------
// Stage 1: AdderNet conv (L1 distance) + residual, per-block BN partial stats.
// One block = one (n, 16x16 tile). 256 threads = 8 wave32s, thread == pixel.
// x tile (64ch x 18x18 halo) pulled into LDS with GLOBAL_LOAD_ASYNC_TO_LDS
// (ASYNCcnt path, no VGPR round-trip); each x value is then reused 576x.
// Weight indices are uniform -> scalar loads (SGPR operand into VALU).
// ---------------------------------------------------------------------------
__global__ __launch_bounds__(256) void adder_stage1(
    const float* __restrict__ x, const float* __restrict__ w,
    float* __restrict__ y, float* __restrict__ partials)
{
  __shared__ float sX[CIN * HALO * HALO];   // ~81 KB of the WGP's 320 KB LDS
  __shared__ float sStats[2 * COUT];        // [0..63]=sum, [64..127]=sumsq

  const int tid = threadIdx.x;
  const int b   = blockIdx.x;
  const int n   = b >> 6;
  const int t   = b & 63;
  const int h0  = (t >> 3) * TILE;
  const int w0  = (t & 7) * TILE;

  if (tid < 2 * COUT) sStats[tid] = 0.0f;

  // Async global->LDS fill of the haloed x tile (zero pad at image borders).
  const float* xn = x + (size_t)n * CIN * HH * WW;
  for (int i = tid; i < CIN * HALO * HALO; i += 256) {
    int ci = i / (HALO * HALO);
    int r  = i - ci * (HALO * HALO);
    int lh = r / HALO, lw = r - lh * HALO;
    int gh = h0 + lh - 1, gw = w0 + lw - 1;
    unsigned lds_off = (unsigned)(unsigned long long)(lds_float*)&sX[i];
    if (gh >= 0 && gh < HH && gw >= 0 && gw < WW) {
      unsigned goff = (unsigned)(((ci * HH + gh) * WW + gw) * 4);
      // GVS mode: mem = SGPR base + per-lane 32-bit offset; dest = LDS offset.
      asm volatile("global_load_async_to_lds_b32 %0, %1, %2"
                   :: "v"(lds_off), "v"(goff), "s"(xn) : "memory");
    } else {
      sX[i] = 0.0f;    // halo outside the image
    }
  }
  asm volatile("s_wait_asynccnt 0x0" ::: "memory");
  __syncthreads();

  const int ty = tid >> 4, tx = tid & 15;
  const int lane = tid & 31;

  #pragma unroll 1
  for (int cog = 0; cog < COUT / 4; ++cog) {
    // Prefetch next channel-group's weights into cache (global_prefetch_b8).
    __builtin_prefetch(&w[(size_t)(((cog + 1) & 15) * 4) * WCO], 0, 1);

    float acc0 = 0.f, acc1 = 0.f, acc2 = 0.f, acc3 = 0.f;
    #pragma unroll 1
    for (int ci = 0; ci < CIN; ++ci) {
      float xv[9];
      #pragma unroll
      for (int kh = 0; kh < 3; ++kh)
        #pragma unroll
        for (int kw = 0; kw < 3; ++kw)
          xv[kh * 3 + kw] = sX[(ci * HALO + ty + kh) * HALO + tx + kw];

      const float* wp = w + ((size_t)(cog * 4) * CIN + ci) * 9;  // uniform -> s_load
      #pragma unroll
      for (int k = 0; k < 9; ++k) {
        acc0 += fabsf(xv[k] - wp[k]);
        acc1 += fabsf(xv[k] - wp[WCO + k]);
        acc2 += fabsf(xv[k] - wp[2 * WCO + k]);
        acc3 += fabsf(xv[k] - wp[3 * WCO + k]);
      }
    }

    float accs[4] = {acc0, acc1, acc2, acc3};
    #pragma unroll
    for (int u = 0; u < 4; ++u) {
      const int co = cog * 4 + u;
      // residual: y = x - L1sum  (out-channel co aliases in-channel co, C==CO)
      const float xc = sX[(co * HALO + ty + 1) * HALO + tx + 1];
      const float yv = xc - accs[u];
      y[(((size_t)n * COUT + co) * HH + (h0 + ty)) * WW + (w0 + tx)] = yv;

      // wave32 butterfly reduction of (sum, sumsq), then LDS atomic per wave
      float s = yv, q = yv * yv;
      #pragma unroll
      for (int off = 16; off >= 1; off >>= 1) {
        s += __shfl_xor(s, off, 32);
        q += __shfl_xor(q, off, 32);
      }
      if (lane == 0) {
        atomicAdd(&sStats[co], s);
        atomicAdd(&sStats[COUT + co], q);
      }
    }
  }
  __syncthreads();
  if (tid < 2 * COUT)
    partials[(size_t)blockIdx.x * (2 * COUT) + tid] = sStats[tid];
}

// ---------------------------------------------------------------------------
// Stage 2: reduce 512 block-partials -> per-channel BN scale/shift, using the
// matrix pipe: V_WMMA_F32_16X16X4_F32 with B == ones performs exact fp32 row
// sums (C += A x 1). One wave (EXEC all ones, as WMMA requires).
// A layout (16x4 f32): lane l holds row m=l&15, K = 2*(l>>4) .. 2*(l>>4)+1.
// C layout: lanes 0-15 VGPR r -> row r; lanes 16-31 VGPR r -> row 8+r.
// ---------------------------------------------------------------------------
__global__ __launch_bounds__(32) void stats_stage2(
    const float* __restrict__ partials, const float* __restrict__ gamma,
    const float* __restrict__ beta, float* __restrict__ finals)
{
  __shared__ float sstats[2 * COUT];
  const int l  = threadIdx.x;       // 0..31, one full wave32
  const int m  = l & 15;
  const int k0 = (l >> 4) * 2;

  v2f bones; bones.x = 1.0f; bones.y = 1.0f;

  for (int jb = 0; jb < 2 * COUT; jb += 16) {
    v8f c = {};
    for (int kb = 0; kb < NBLK1; kb += 4) {
      v2f a;
      a.x = partials[(size_t)(kb + k0)     * (2 * COUT) + jb + m];
      a.y = partials[(size_t)(kb + k0 + 1) * (2 * COUT) + jb + m];
      c = __builtin_amdgcn_wmma_f32_16x16x4_f32(
          /*neg_a=*/false, a, /*neg_b=*/false, bones,
          /*c_mod=*/(short)0, c, /*reuse_a=*/false, /*reuse_b=*/false);
    }
    if (l == 0) {                    // rows 0..7 (N=0 column)
      #pragma unroll
      for (int r = 0; r < 8; ++r) sstats[jb + r] = c[r];
    } else if (l == 16) {            // rows 8..15 (N=0 column)
      #pragma unroll
      for (int r = 0; r < 8; ++r) sstats[jb + 8 + r] = c[r];
    }
  }
  __syncthreads();

  #pragma unroll
  for (int t = 0; t < 2; ++t) {
    const int co = l + 32 * t;
    const float mean = sstats[co] / CNTF;
    const float var  = sstats[COUT + co] / CNTF - mean * mean;
    const float inv  = rsqrtf(var + 1e-5f);
    const float scale = gamma[co] * inv;
    const float shift = beta[co] - mean * scale;
    finals[co] = scale;
    finals[COUT + co] = shift;
  }
}

// ---------------------------------------------------------------------------
// Stage 3: normalize + affine + PowerActivation: sign(v)*(|v|+1e-12)^alpha.
// Pure bandwidth: 32 MB in, 32 MB out -> float4 (b128) per thread.
// ---------------------------------------------------------------------------
__global__ __launch_bounds__(256) void act_stage3(
    const float4* __restrict__ y4, const float* __restrict__ finals,
    const float* __restrict__ alpha_p, float4* __restrict__ out4)
{
  const int idx4 = blockIdx.x * 256 + threadIdx.x;     // 4 elements per thread
  const float alpha = alpha_p[0];
  // 4 consecutive elements share a channel (H*W = 16384 is 4-aligned)
  const int co = ((idx4 * 4) >> 14) & (COUT - 1);
  const float scale = finals[co];
  const float shift = finals[COUT + co];

  float4 yv = y4[idx4];
  float r[4] = {yv.x, yv.y, yv.z, yv.w};
  #pragma unroll
  for (int k = 0; k < 4; ++k) {
    const float v = r[k] * scale + shift;
    const float p = __powf(fabsf(v) + 1e-12f, alpha);
    r[k] = (v > 0.f) ? p : ((v < 0.f) ? -p : 0.f);
  }
  out4[idx4] = make_float4(r[0], r[1], r[2], r[3]);
}

// ---------------------------------------------------------------------------
extern "C" void kernel_launch(void* const* d_in, const int* in_sizes, int n_in,
                              void* d_out, int out_size, void* d_ws, size_t ws_size,
                              hipStream_t stream) {
  const float* x     = (const float*)d_in[0];
  const float* w     = (const float*)d_in[1];
  const float* gamma = (const float*)d_in[2];
  const float* beta  = (const float*)d_in[3];
  const float* alpha = (const float*)d_in[4];

  float* ws       = (float*)d_ws;
  float* y        = ws;                         // NPIX floats (32 MB)
  float* partials = y + NPIX;                   // NBLK1 * 128 floats (256 KB)
  float* finals   = partials + NBLK1 * 2 * COUT;  // 128 floats (scale|shift)

  adder_stage1<<<NBLK1, 256, 0, stream>>>(x, w, y, partials);
  stats_stage2<<<1, 32, 0, stream>>>(partials, gamma, beta, finals);
  act_stage3<<<NPIX / 1024, 256, 0, stream>>>(
      (const float4*)y, finals, alpha, (float4*)d_out);
}